// _StratifiedRegTsneModel_14181982011717
// MI455X (gfx1250) — compile-verified
//
#include <hip/hip_runtime.h>
#include <hip/hip_bf16.h>

typedef __attribute__((ext_vector_type(2))) float v2f;
typedef __attribute__((ext_vector_type(8))) float v8f;

// Only meaningful during the device pass; host pass doesn't see amdgcn
// builtins via __has_builtin even though it can parse calls to them.
#if defined(__HIP_DEVICE_COMPILE__) && !__has_builtin(__builtin_amdgcn_wmma_f32_16x16x4_f32)
#error "no __builtin_amdgcn_wmma_f32_16x16x4_f32 on this toolchain (device pass)"
#endif

constexpr int   NB   = 2;
constexpr int   NN   = 4096;
constexpr int   FF   = 128;
constexpr float EPSV = 1e-12f;
constexpr int   TILE = 64;           // super-tile edge handled by one block
constexpr int   LSTR = FF + 4;       // LDS row stride (floats), pad vs bank conflicts

// ---------------------------------------------------------------------------
// Prep: Y = X * W (row-scale by W[f]), sq[row] = ||Y_row||^2.  One wave per row.
// ---------------------------------------------------------------------------
__global__ __launch_bounds__(256) void prep_kernel(
    const float* __restrict__ X, const float* __restrict__ W,
    float* __restrict__ Yw, float* __restrict__ sq)
{
    const int wave = threadIdx.x >> 5;
    const int lane = threadIdx.x & 31;
    const int row  = blockIdx.x * 8 + wave;          // row in [0, NB*NN)
    const float4 w4 = *(const float4*)(W + lane * 4);
    const float4 x4 = *(const float4*)(X + (size_t)row * FF + lane * 4);
    float4 y4;
    y4.x = x4.x * w4.x; y4.y = x4.y * w4.y;
    y4.z = x4.z * w4.z; y4.w = x4.w * w4.w;
    *(float4*)(Yw + (size_t)row * FF + lane * 4) = y4;
    float s = y4.x * y4.x + y4.y * y4.y + y4.z * y4.z + y4.w * y4.w;
    #pragma unroll
    for (int off = 16; off; off >>= 1) s += __shfl_xor(s, off, 32);
    if (lane == 0) sq[row] = s;
}

// ---------------------------------------------------------------------------
// Tile kernel: 512 threads = 16 waves; block owns a 64x64 tile of the NxN
// matrix for batch blockIdx.z; wave (ti,tj) computes one 16x16 WMMA tile.
// PASS 1: accumulate sum of t.   PASS 2: accumulate KL terms.
// ---------------------------------------------------------------------------
template <int PASS>
__global__ __launch_bounds__(512) void tsne_tile_kernel(
    const float* __restrict__ Yw,   const float* __restrict__ sq,
    const float* __restrict__ beta, const float* __restrict__ P,
    const float* __restrict__ sumT, float* __restrict__ partials)
{
    __shared__ __align__(16) float Yi[TILE][LSTR];
    __shared__ __align__(16) float Yj[TILE][LSTR];
    __shared__ float sqi[TILE], sqj[TILE], bti[TILE], btj[TILE];
    __shared__ float red[16];

    const int b   = blockIdx.z;
    const int i0  = blockIdx.y * TILE;
    const int j0  = blockIdx.x * TILE;
    const int tid = threadIdx.x;
    const float* Yb = Yw + (size_t)b * NN * FF;

    // Stage 64x128 panels (2048 float4 per side; 4 per thread per side).
    #pragma unroll
    for (int it = 0; it < 4; ++it) {
        const int idx = tid + it * 512;      // float4 index
        const int r   = idx >> 5;            // FF/4 = 32 float4 per row
        const int c   = (idx & 31) << 2;     // float column
        *(float4*)(&Yi[r][c]) = *(const float4*)(Yb + (size_t)(i0 + r) * FF + c);
        *(float4*)(&Yj[r][c]) = *(const float4*)(Yb + (size_t)(j0 + r) * FF + c);
    }
    if (tid < TILE) {
        sqi[tid] = sq[b * NN + i0 + tid];
        bti[tid] = beta[b * NN + i0 + tid];
    } else if (tid < 2 * TILE) {
        const int t = tid - TILE;
        sqj[t] = sq[b * NN + j0 + t];
        btj[t] = beta[b * NN + j0 + t];
    }
    __syncthreads();

    const int wave = tid >> 5;
    const int lane = tid & 31;
    const int half = lane >> 4;
    const int r16  = lane & 15;
    const int ti   = wave >> 2;   // 0..3
    const int tj   = wave & 3;    // 0..3

    // f32 16x16x4 WMMA frag layout: lanes 0-15 -> K pair {k,k+1},
    // lanes 16-31 -> {k+2,k+3}; A rows / B cols indexed by lane%16.
    const float* Ai = &Yi[ti * 16 + r16][2 * half];
    const float* Bj = &Yj[tj * 16 + r16][2 * half];

    v8f acc = {};
    #pragma unroll
    for (int k = 0; k < FF; k += 4) {
        v2f a = *(const v2f*)(Ai + k);
        v2f bb = *(const v2f*)(Bj + k);
        acc = __builtin_amdgcn_wmma_f32_16x16x4_f32(
            false, a, false, bb, (short)0, acc, false, false);
    }

    // C/D layout: lane holds column n = lane%16; VGPR v -> row v + 8*(lane/16).
    const int   gj   = j0 + tj * 16 + r16;
    const float sqjv = sqj[tj * 16 + r16];
    const float btjv = btj[tj * 16 + r16];

    float invS = 0.0f;
    if (PASS == 2) invS = 1.0f / sumT[b];

    float local = 0.0f;
    #pragma unroll
    for (int v = 0; v < 8; ++v) {
        const int   mi = ti * 16 + 8 * half + v;   // local row in [0,64)
        const int   gi = i0 + mi;
        const float g  = acc[v];
        const float d2 = fmaxf(sqi[mi] + sqjv - 2.0f * g, 0.0f);
        const float s  = 0.5f * (bti[mi] + btjv);
        const float t  = 1.0f / (1.0f + s * d2);   // t-kernel
        if (PASS == 1) {
            local += (gi == gj) ? 0.0f : t;
        } else {
            if (gi != gj) {
                const float p = P[(size_t)b * NN * NN + (size_t)gi * NN + gj];
                const float Q = fmaxf(t * invS, EPSV);
                local += p * (__logf(p) - __logf(Q));
            }
        }
    }

    // Deterministic block reduction -> one partial per block.
    #pragma unroll
    for (int off = 16; off; off >>= 1) local += __shfl_xor(local, off, 32);
    if (lane == 0) red[wave] = local;
    __syncthreads();
    if (wave == 0) {
        float vsum = (lane < 16) ? red[lane] : 0.0f;
        #pragma unroll
        for (int off = 16; off; off >>= 1) vsum += __shfl_xor(vsum, off, 32);
        if (lane == 0) {
            const int blk = (blockIdx.z * gridDim.y + blockIdx.y) * gridDim.x + blockIdx.x;
            partials[blk] = vsum;
        }
    }
}

// ---------------------------------------------------------------------------
// Deterministic reductions (fixed-order stride loop + shared tree).
// ---------------------------------------------------------------------------
__global__ __launch_bounds__(256) void reduce_sumT_kernel(
    const float* __restrict__ partials, float* __restrict__ sumT)
{
    __shared__ float sm[256];
    float s = 0.0f;
    const int base = blockIdx.x * 4096;
    for (int i = threadIdx.x; i < 4096; i += 256) s += partials[base + i];
    sm[threadIdx.x] = s;
    __syncthreads();
    for (int off = 128; off; off >>= 1) {
        if (threadIdx.x < off) sm[threadIdx.x] += sm[threadIdx.x + off];
        __syncthreads();
    }
    if (threadIdx.x == 0) sumT[blockIdx.x] = sm[0];
}

__global__ __launch_bounds__(256) void reduce_out_kernel(
    const float* __restrict__ partials, float* __restrict__ out)
{
    __shared__ float sm[256];
    float s = 0.0f;
    for (int i = threadIdx.x; i < NB * 4096; i += 256) s += partials[i];
    sm[threadIdx.x] = s;
    __syncthreads();
    for (int off = 128; off; off >>= 1) {
        if (threadIdx.x < off) sm[threadIdx.x] += sm[threadIdx.x + off];
        __syncthreads();
    }
    if (threadIdx.x == 0) out[0] = sm[0] * (1.0f / NB);
}

// ---------------------------------------------------------------------------
extern "C" void kernel_launch(void* const* d_in, const int* in_sizes, int n_in,
                              void* d_out, int out_size, void* d_ws, size_t ws_size,
                              hipStream_t stream)
{
    (void)in_sizes; (void)n_in; (void)out_size; (void)ws_size;
    const float* X    = (const float*)d_in[0];   // [B,N,F]
    const float* P    = (const float*)d_in[1];   // [B,N,N]
    const float* beta = (const float*)d_in[2];   // [B,N,1]
    const float* W    = (const float*)d_in[3];   // [1,F]

    float* ws   = (float*)d_ws;
    float* Yw   = ws;                            // NB*NN*FF
    float* sq   = Yw + (size_t)NB * NN * FF;     // NB*NN
    float* p1   = sq + NB * NN;                  // NB*4096 pass-1 partials
    float* sumT = p1 + NB * 4096;                // NB
    float* p2   = sumT + NB;                     // NB*4096 pass-2 partials

    prep_kernel<<<NB * NN / 8, 256, 0, stream>>>(X, W, Yw, sq);

    dim3 grid(NN / TILE, NN / TILE, NB);         // 64 x 64 x 2
    tsne_tile_kernel<1><<<grid, 512, 0, stream>>>(Yw, sq, beta, nullptr, nullptr, p1);
    reduce_sumT_kernel<<<NB, 256, 0, stream>>>(p1, sumT);
    tsne_tile_kernel<2><<<grid, 512, 0, stream>>>(Yw, sq, beta, P, sumT, p2);
    reduce_out_kernel<<<1, 256, 0, stream>>>(p2, (float*)d_out);
}